// CrossViewAttention_55405078119180
// MI455X (gfx1250) — compile-verified
//
#include <hip/hip_runtime.h>

typedef float v2f __attribute__((ext_vector_type(2)));
typedef float v8f __attribute__((ext_vector_type(8)));

#define B_DIM 2
#define C_DIM 128
#define S_DIM 64
#define M_DIM 4096   // S*S
#define H_DIM 128
#define W_DIM 512
#define MID   32
#define EPS   1e-5f

// -----------------------------------------------------------------------------
// Fused LayerNorm (over C=128) + GEMM [rows x 128] @ [128 x 32] (+bias) using
// V_WMMA_F32_16X16X4_F32. Input element (b, c, r) lives at in[b*C*RS + c*RS + r]
// (channel-major, rows contiguous -> coalesced when lanes stride over rows).
// Output row (b*RS + r) is 32 floats, contiguous.
//
// One wave computes one 16-row x 32-col output tile; 4 waves / block.
// Single global pass: half-waves split the c-range, stage the 16x128 tile into
// per-wave LDS while accumulating LN stats, combine stats via shfl_xor(16),
// then run the WMMA K-loop entirely out of LDS.
// -----------------------------------------------------------------------------
__global__ void __launch_bounds__(128) ln_proj_wmma(
    const float* __restrict__ in, float* __restrict__ out,
    const float* __restrict__ gamma, const float* __restrict__ beta,
    const float* __restrict__ Wmat, const float* __restrict__ bias, int RS)
{
  __shared__ float sW[C_DIM * MID];       // 16 KB projection weights
  __shared__ float sG[C_DIM];
  __shared__ float sBe[C_DIM];
  __shared__ float sBias[MID];
  __shared__ float sX[4][C_DIM * 16];     // 32 KB: per-wave input tiles [c][row]

  const int tid = threadIdx.x;
  for (int i = tid; i < C_DIM * MID; i += 128) sW[i] = Wmat[i];
  if (tid < C_DIM) { sG[tid] = gamma[tid]; sBe[tid] = beta[tid]; }
  if (tid < MID)   sBias[tid] = bias[tid];

  const int lane  = tid & 31;
  const int wv    = tid >> 5;
  const int tile  = blockIdx.x * 4 + wv;
  const int grow0 = tile * 16;              // global output row of this tile
  const int bidx  = grow0 / RS;             // batch (RS multiple of 16)
  const int r0    = grow0 % RS;
  const size_t bbase = (size_t)bidx * C_DIM * RS;
  const int    row   = lane & 15;           // this lane's tile row (== A-row M)
  const size_t rofs  = (size_t)(r0 + row);
  float* xt = &sX[wv][0];

  // --- Single-pass stage + LN stats: lanes 0-15 cover c=[0,64), 16-31 c=[64,128)
  const int cbase = (lane >> 4) * 64;
  float s1 = 0.f, s2 = 0.f;
  for (int i = 0; i < 64; ++i) {
    const int c = cbase + i;
    // speculative prefetch down the strided column (lowers to global_prefetch_b8)
    const int cp = (i < 48) ? c + 16 : c;
    __builtin_prefetch(&in[bbase + (size_t)cp * RS + rofs], 0, 1);
    const float v = in[bbase + (size_t)c * RS + rofs];
    xt[c * 16 + row] = v;
    s1 += v; s2 += v * v;
  }
  // combine half-wave partial sums -> full row statistics on every lane
  s1 += __shfl_xor(s1, 16, 32);
  s2 += __shfl_xor(s2, 16, 32);
  const float mean = s1 * (1.0f / C_DIM);
  const float var  = s2 * (1.0f / C_DIM) - mean * mean;
  const float rstd = rsqrtf(var + EPS);

  __syncthreads();   // weights staged + cross-lane LDS tile visibility

  // --- WMMA accumulation over K = C in steps of 4 (f32 WMMA shape 16x16x4).
  // A layout (ISA 7.12.2): lanes 0-15 -> M=lane, K-pair {0,1}; lanes 16-31 ->
  // M=lane-16, K-pair {2,3}. B layout: VGPR0 = K rows {0,2}, VGPR1 = {1,3},
  // N striped across half-wave lanes. A-reads hit banks 0-15 / 32-47: no conflict.
  v8f acc0 = {};  // N = 0..15
  v8f acc1 = {};  // N = 16..31
  const int khalf = (lane >> 4) << 1;   // 0 or 2
  const int nloc  = lane & 15;
  for (int kk = 0; kk < C_DIM / 4; ++kk) {
    const int c0 = kk * 4 + khalf;
    const float x0 = xt[c0 * 16 + row];
    const float x1 = xt[(c0 + 1) * 16 + row];
    v2f a;
    a.x = (x0 - mean) * rstd * sG[c0]     + sBe[c0];
    a.y = (x1 - mean) * rstd * sG[c0 + 1] + sBe[c0 + 1];
    v2f b0, b1;
    b0.x = sW[c0 * MID + nloc];            b0.y = sW[(c0 + 1) * MID + nloc];
    b1.x = sW[c0 * MID + 16 + nloc];       b1.y = sW[(c0 + 1) * MID + 16 + nloc];
    acc0 = __builtin_amdgcn_wmma_f32_16x16x4_f32(false, a, false, b0,
                                                 (short)0, acc0, false, false);
    acc1 = __builtin_amdgcn_wmma_f32_16x16x4_f32(false, a, false, b1,
                                                 (short)0, acc1, false, false);
  }

  // D layout: VGPR r -> (M=r, N=lane) for lanes 0-15, (M=r+8, N=lane-16) else.
  const int rowAdd = (lane >> 4) * 8;
  for (int rr = 0; rr < 8; ++rr) {
    const int orow = grow0 + rr + rowAdd;
    out[(size_t)orow * MID + nloc]      = acc0[rr] + sBias[nloc];
    out[(size_t)orow * MID + 16 + nloc] = acc1[rr] + sBias[16 + nloc];
  }
}

// -----------------------------------------------------------------------------
// Attention: one wave32 per (b,m). Lane l owns h = l + 32*j, j=0..3.
// k_all row for (b,h,u[b,m]) is 32 contiguous floats (128B) -> float4 loads,
// L2-resident (16 MB). Two-pass softmax over H=128 via wave32 xor-reductions.
// -----------------------------------------------------------------------------
__global__ void __launch_bounds__(128) attn_softmax(
    const float* __restrict__ q, const float* __restrict__ kall,
    const float* __restrict__ height, const int* __restrict__ u,
    float* __restrict__ out)
{
  const int lane = threadIdx.x & 31;
  const int bm   = blockIdx.x * 4 + (threadIdx.x >> 5);
  const int b    = bm >> 12;            // / M_DIM
  const int mm   = bm & (M_DIM - 1);

  int wi = u[(size_t)b * M_DIM + mm];
  wi = wi < 0 ? 0 : (wi > W_DIM - 1 ? W_DIM - 1 : wi);

  const float4* q4 = (const float4*)(q + (size_t)bm * MID);
  float4 qa[8];
#pragma unroll
  for (int t = 0; t < 8; ++t) qa[t] = q4[t];   // same 128B for all lanes -> broadcast

  const float scale = 0.17677669529663687f;    // 32^-0.5
  float d[4], vv[4];
#pragma unroll
  for (int j = 0; j < 4; ++j) {
    const int h = lane + 32 * j;
    const size_t hw = (size_t)(b * H_DIM + h) * W_DIM + wi;
    const float4* k4 = (const float4*)(kall + hw * MID);
    float acc = 0.f;
#pragma unroll
    for (int t = 0; t < 8; ++t) {
      const float4 kv = k4[t];
      acc += qa[t].x * kv.x + qa[t].y * kv.y + qa[t].z * kv.z + qa[t].w * kv.w;
    }
    d[j]  = acc * scale;
    vv[j] = height[hw];
  }

  float mx = fmaxf(fmaxf(d[0], d[1]), fmaxf(d[2], d[3]));
#pragma unroll
  for (int off = 16; off; off >>= 1) mx = fmaxf(mx, __shfl_xor(mx, off, 32));

  float num = 0.f, den = 0.f;
#pragma unroll
  for (int j = 0; j < 4; ++j) {
    const float e = __expf(d[j] - mx);
    den += e; num += e * vv[j];
  }
#pragma unroll
  for (int off = 16; off; off >>= 1) {
    num += __shfl_xor(num, off, 32);
    den += __shfl_xor(den, off, 32);
  }
  if (lane == 0) out[bm] = num / den;
}

extern "C" void kernel_launch(void* const* d_in, const int* in_sizes, int n_in,
                              void* d_out, int out_size, void* d_ws, size_t ws_size,
                              hipStream_t stream) {
  (void)in_sizes; (void)n_in; (void)out_size; (void)ws_size;
  const float* sat_x  = (const float*)d_in[0];
  const float* grd_x  = (const float*)d_in[1];
  const float* grd_h  = (const float*)d_in[2];
  const int*   u      = (const int*)d_in[3];
  const float* ln_q_g = (const float*)d_in[4];
  const float* ln_q_b = (const float*)d_in[5];
  const float* wq     = (const float*)d_in[6];
  const float* bq     = (const float*)d_in[7];
  const float* ln_k_g = (const float*)d_in[8];
  const float* ln_k_b = (const float*)d_in[9];
  const float* wk     = (const float*)d_in[10];
  const float* bk     = (const float*)d_in[11];
  float* out = (float*)d_out;

  float* q_ws = (float*)d_ws;                              // 8192*32 f32 = 1 MB
  float* k_ws = q_ws + (size_t)B_DIM * M_DIM * MID;        // 131072*32 f32 = 16 MB

  // q = LN(sat_x^T) @ wq + bq   (rows = B*M, RS = M)
  ln_proj_wmma<<<(B_DIM * M_DIM) / 64, 128, 0, stream>>>(
      sat_x, q_ws, ln_q_g, ln_q_b, wq, bq, M_DIM);
  // k_all = LN(grd_x^T) @ wk + bk for ALL (h,w)  (rows = B*H*W, RS = H*W)
  ln_proj_wmma<<<(B_DIM * H_DIM * W_DIM) / 64, 128, 0, stream>>>(
      grd_x, k_ws, ln_k_g, ln_k_b, wk, bk, H_DIM * W_DIM);
  // softmax attention + height reduction
  attn_softmax<<<(B_DIM * M_DIM) / 4, 128, 0, stream>>>(
      q_ws, k_ws, grd_h, u, out);
}